// MXQuantizer_90692529423087
// MI455X (gfx1250) — compile-verified
//
#include <hip/hip_runtime.h>

// MX (block-32, E4M3-style: EBITS=4, MBITS=3, EMAX=8, max ±448) quant-dequant
// along axis 1 of x[8][4096][2048] (row-major). A block = 32 consecutive rows
// of one column -> strided by 8KB in memory. Strategy: stage 32x256 tiles in
// LDS via CDNA5 async global->LDS copies (contiguous 512B per wave-instr),
// each thread then owns one full MX block (its column) in LDS, quantizes in
// registers, writes back to LDS, and the tile is streamed out with async
// LDS->global stores. Memory-bound: 512MB total @ 23.3TB/s ~= 22us floor.
// Non-temporal hints on both directions: 512MB of once-touched data must not
// thrash the 192MB L2.

#define M_DIM   4096
#define K_DIM   2048
#define BLK_M   32      // MX block size (rows per tile)
#define TILE_K  256     // columns per tile == threads per block
#define NTHREADS 256

__device__ __forceinline__ float exp2i(int e) {
  // exact 2^e for e in [-126, 127]
  return __uint_as_float((unsigned)(e + 127) << 23);
}

__global__ __launch_bounds__(NTHREADS)
void mx_qdq_kernel(const float* __restrict__ x, float* __restrict__ out) {
  __shared__ float tile[BLK_M * TILE_K];   // 32 KB

  const int t  = threadIdx.x;
  const int b  = blockIdx.z;
  const int m0 = blockIdx.y * BLK_M;
  const int k0 = blockIdx.x * TILE_K;

  // Low 32 bits of a generic LDS pointer are the wave-relative LDS byte offset
  // (LDS aperture lives entirely in the high dword).
  const unsigned ldsBase = (unsigned)(size_t)(&tile[0]);

  // ---- Tile mapping for async copies: round j, thread t moves 16 bytes of
  // row r = j*4 + t/64 at columns (t%64)*4 .. +3. Each wave-instruction is a
  // contiguous 512B global segment. 8 rounds cover the 32x256 tile. ----
  const int r_in = t >> 6;             // 0..3
  const int c4   = (t & 63) << 2;      // 0,4,...,252
  const unsigned gbase =
      (((unsigned)(b * M_DIM + m0 + r_in)) * (unsigned)K_DIM +
       (unsigned)(k0 + c4)) * 4u;                       // byte offset < 2^31
  const unsigned lbase = ldsBase + ((unsigned)(r_in * TILE_K + c4)) * 4u;
  const unsigned gRowStep = 4u * (unsigned)K_DIM * 4u;  // 4 rows in bytes
  const unsigned lRowStep = 4u * (unsigned)TILE_K * 4u;

  // ---- async load global -> LDS (ASYNCcnt), non-temporal: streamed once ----
  #pragma unroll
  for (int j = 0; j < 8; ++j) {
    unsigned goff = gbase + (unsigned)j * gRowStep;
    unsigned loff = lbase + (unsigned)j * lRowStep;
    asm volatile("global_load_async_to_lds_b128 %0, %1, %2 th:TH_LOAD_NT"
                 :: "v"(loff), "v"(goff), "s"(x)
                 : "memory");
  }
  asm volatile("s_wait_asynccnt 0" ::: "memory");
  __syncthreads();

  // ---- thread t owns one MX block: column k0+t, rows m0..m0+31 ----
  float v[BLK_M];
  float maxabs = 0.0f;
  #pragma unroll
  for (int i = 0; i < BLK_M; ++i) {
    v[i] = tile[i * TILE_K + t];       // ds_load_b32, bank-conflict-free
    maxabs = fmaxf(maxabs, fabsf(v[i]));
  }

  const float safe = (maxabs > 0.0f) ? maxabs : 1.0f;
  // shared_exp = floor(log2(safe)) - EMAX ; exact for normals via exponent bits
  const int se = (int)((__float_as_uint(safe) >> 23) & 0xFF) - 127 - 8;
  const float scale     = ldexpf(1.0f, se);    // handles subnormal edge
  const float inv_scale = ldexpf(1.0f, -se);

  #pragma unroll
  for (int i = 0; i < BLK_M; ++i) {
    float vs  = v[i] * inv_scale;              // exact: power-of-two scaling
    float mag = fabsf(vs);
    int e = (int)((__float_as_uint(mag) >> 23) & 0xFF) - 127;
    e = (e < -6) ? -6 : e;                     // EMIN clamp (covers denorm/0)
    float r   = exp2i(3 - e);                  // 1/ulp, e in [-6,8] -> normal
    float ulp = exp2i(e - 3);
    float q = rintf(vs * r) * ulp;             // RTE == jnp.round half-even
    q = fminf(fmaxf(q, -448.0f), 448.0f);
    tile[i * TILE_K + t] = q * scale;          // zero block stays zero
  }
  __syncthreads();

  // ---- async store LDS -> global (ASYNCcnt), non-temporal ----
  #pragma unroll
  for (int j = 0; j < 8; ++j) {
    unsigned goff = gbase + (unsigned)j * gRowStep;
    unsigned loff = lbase + (unsigned)j * lRowStep;
    asm volatile("global_store_async_from_lds_b128 %0, %1, %2 th:TH_STORE_NT"
                 :: "v"(goff), "v"(loff), "s"(out)
                 : "memory");
  }
  asm volatile("s_wait_asynccnt 0" ::: "memory");
}

extern "C" void kernel_launch(void* const* d_in, const int* in_sizes, int n_in,
                              void* d_out, int out_size, void* d_ws, size_t ws_size,
                              hipStream_t stream) {
  (void)n_in; (void)d_ws; (void)ws_size; (void)out_size;
  const float* x = (const float*)d_in[0];
  float* out = (float*)d_out;
  const int batches = in_sizes[0] / (M_DIM * K_DIM);   // 8 for the reference
  dim3 grid(K_DIM / TILE_K, M_DIM / BLK_M, batches);
  dim3 block(NTHREADS, 1, 1);
  mx_qdq_kernel<<<grid, block, 0, stream>>>(x, out);
}